// SpectrumGenerator_32444182954512
// MI455X (gfx1250) — compile-verified
//
#include <hip/hip_runtime.h>
#include <math.h>

typedef float v2f __attribute__((ext_vector_type(2)));
typedef float v8f __attribute__((ext_vector_type(8)));

#define MBATCH 1024
#define NOBS   8192
#define NEG_SLOPE 0.2f

#define BM 64
#define BN 128
#define BK 16
#define LDR 20   // padded LDS row stride (floats): 80B = 16B-aligned, conflict-free b64 reads

// ---------------------------------------------------------------------------
// GEMM:  Y[M,N] = act( X[M,K] * W[N,K]^T + b[N] ),  fp32 via V_WMMA_F32_16X16X4_F32
// 256 threads (8 waves), tile 64(M) x 128(N), K-step 16, double-buffered LDS.
// Wave w: M-subtile (w>>1)*16, four N-subtiles at (w&1)*64 + {0,16,32,48}.
// LDS layout [row][k]: fragments are single ds_load_b64, staging ds_store_b128.
// B tail rows are CLAMPED (not guarded): garbage only reaches accumulator
// columns with gn >= N, which the epilogue never stores -> keeps b128 loads.
// ---------------------------------------------------------------------------
template<bool ACT>
__global__ __launch_bounds__(256)
void gemm_wmma(const float* __restrict__ X, const float* __restrict__ W,
               const float* __restrict__ b, float* __restrict__ Y,
               int M, int N, int K)
{
    __shared__ float As[2][BM][LDR];
    __shared__ float Bs[2][BN][LDR];

    const int t    = threadIdx.x;
    const int lane = t & 31;
    const int wave = t >> 5;
    const int l16  = lane & 15;
    const int khi  = (lane >> 4) << 1;   // 0 or 2
    const int mhi  = (lane >> 4) << 3;   // 0 or 8 (C/D layout)

    const int mBase  = (wave >> 1) * 16;   // 0..48
    const int nGroup = (wave & 1) * 64;    // 0 or 64

    const int m0 = blockIdx.y * BM;
    const int n0 = blockIdx.x * BN;

    const int rrow = t >> 2;          // 0..63
    const int rcol = (t & 3) << 2;    // 0,4,8,12

    v8f acc[4] = {v8f{}, v8f{}, v8f{}, v8f{}};

    // ---- helpers -----------------------------------------------------------
    auto fetchA = [&](int k0) -> float4 {
        return *(const float4*)(X + (size_t)(m0 + rrow) * K + (k0 + rcol));
    };
    auto fetchB = [&](int k0, int rowoff) -> float4 {
        int n = n0 + rowoff + rrow;
        n = (n < N) ? n : (N - 1);                      // clamp, stay in-bounds
        return *(const float4*)(W + (size_t)n * K + (k0 + rcol));
    };
    auto stage = [&](int buf, float4 ra, float4 rb0, float4 rb1) {
        *(float4*)&As[buf][rrow][rcol]      = ra;
        *(float4*)&Bs[buf][rrow][rcol]      = rb0;
        *(float4*)&Bs[buf][64 + rrow][rcol] = rb1;
    };
    auto compute = [&](int buf) {
        v2f a[4], bf[4][4];
        #pragma unroll
        for (int kq = 0; kq < 4; ++kq)
            a[kq] = *(const v2f*)&As[buf][mBase + l16][4 * kq + khi];
        #pragma unroll
        for (int kq = 0; kq < 4; ++kq)
            #pragma unroll
            for (int j = 0; j < 4; ++j)
                bf[kq][j] = *(const v2f*)&Bs[buf][nGroup + j * 16 + l16][4 * kq + khi];
        #pragma unroll
        for (int kq = 0; kq < 4; ++kq)
            #pragma unroll
            for (int j = 0; j < 4; ++j)
                acc[j] = __builtin_amdgcn_wmma_f32_16x16x4_f32(
                             false, a[kq], false, bf[kq][j], (short)0, acc[j],
                             false, false);
    };
    // ------------------------------------------------------------------------

    float4 ra  = fetchA(0);
    float4 rb0 = fetchB(0, 0);
    float4 rb1 = fetchB(0, 64);
    stage(0, ra, rb0, rb1);
    __syncthreads();

    int buf = 0;
    for (int k0 = BK; k0 < K; k0 += BK) {
        ra  = fetchA(k0);          // prefetch next tile while computing current
        rb0 = fetchB(k0, 0);
        rb1 = fetchB(k0, 64);
        compute(buf);
        stage(buf ^ 1, ra, rb0, rb1);
        __syncthreads();
        buf ^= 1;
    }
    compute(buf);

    // ---- epilogue: bias + optional leaky-ReLU, guarded N tail --------------
    #pragma unroll
    for (int j = 0; j < 4; ++j) {
        const int gn = n0 + nGroup + j * 16 + l16;
        if (gn < N) {
            const float bb = b[gn];
            #pragma unroll
            for (int r = 0; r < 8; ++r) {
                const int gm = m0 + mBase + mhi + r;
                float yv = acc[j][r] + bb;
                if (ACT) yv = (yv >= 0.f) ? yv : NEG_SLOPE * yv;
                Y[(size_t)gm * N + gn] = yv;
            }
        }
    }
}

// ---------------------------------------------------------------------------
// Interpolation (analytic bracket + 2 L2-resident spectra reads per point)
// ---------------------------------------------------------------------------
struct SubSeg { double s0; double st; int base; int locA; int locB; };
struct IvInfo {
    float maskLo, maskHi;   // s-PAD, e+PAD
    float gFirst, gLast;    // first grid value >= s, last <= e (fp32)
    int   iFirst, iLast;    // their global spectra indices
    int   nSub;
    SubSeg sub[3];
};
struct InterpParams { IvInfo iv[5]; };

__device__ __forceinline__ float gridval(double s0, double st, int i) {
    return (float)(s0 + (double)i * st);   // identical to np.arange(...).astype(f32)
}

__global__ __launch_bounds__(256)
void interp_kernel(const float* __restrict__ obs, const float* __restrict__ spectra,
                   float* __restrict__ out, int D, InterpParams P)
{
    const long idx = (long)blockIdx.x * blockDim.x + threadIdx.x;
    const long total = (long)MBATCH * NOBS;
    if (idx >= total) return;
    const int   row = (int)(idx >> 13);      // / 8192
    const float o   = obs[idx];
    const float* __restrict__ y = spectra + (size_t)row * D;

    float val = 0.f;
    #pragma unroll
    for (int k = 0; k < 5; ++k) {
        const IvInfo& iv = P.iv[k];
        if (o >= iv.maskLo && o <= iv.maskHi) {
            float r;
            if (o <= iv.gFirst)      r = y[iv.iFirst];        // left clamp (pads are flat)
            else if (o >= iv.gLast)  r = y[iv.iLast];         // right clamp
            else {
                int i0 = iv.iFirst, i1 = iv.iLast;
                float x0 = iv.gFirst, x1 = iv.gLast;
                for (int m = 0; m < iv.nSub; ++m) {
                    const SubSeg& sg = iv.sub[m];
                    const float firstV = gridval(sg.s0, sg.st, sg.locA);
                    if (o < firstV) {                      // in a gap between segments
                        const SubSeg& pv = iv.sub[m-1];
                        i0 = pv.base + pv.locB; x0 = gridval(pv.s0, pv.st, pv.locB);
                        i1 = sg.base + sg.locA; x1 = firstV;
                        break;
                    }
                    const float lastV = gridval(sg.s0, sg.st, sg.locB);
                    if (o <= lastV) {                      // inside this uniform segment
                        int jl = (int)(((double)o - sg.s0) / sg.st);
                        if (jl < sg.locA)     jl = sg.locA;
                        if (jl > sg.locB - 1) jl = sg.locB - 1;
                        while (jl > sg.locA     && o < gridval(sg.s0, sg.st, jl))     --jl;
                        while (jl < sg.locB - 1 && o > gridval(sg.s0, sg.st, jl + 1)) ++jl;
                        i0 = sg.base + jl;     x0 = gridval(sg.s0, sg.st, jl);
                        i1 = i0 + 1;           x1 = gridval(sg.s0, sg.st, jl + 1);
                        break;
                    }
                }
                const float y0 = y[i0], y1 = y[i1];
                r = y0 + (o - x0) * (y1 - y0) / (x1 - x0);
            }
            val = r;
        }
    }
    out[idx] = val;
}

// ---------------------------------------------------------------------------
// Host side: replicate numpy arange grid analytically, build interval tables
// ---------------------------------------------------------------------------
static const double HSEGS[7][3] = {
    {15050.0, 15850.0, 0.2}, {15870.0, 16440.0, 0.2}, {16475.0, 17100.0, 0.2},
    {4700.0, 4930.0, 0.05}, {5630.0, 5880.0, 0.05}, {6420.0, 6800.0, 0.05},
    {7500.0, 7920.0, 0.05}
};
static const double HIV[5][2] = {
    {4711.0, 4906.0}, {5647.0, 5875.0}, {6475.0, 6737.0}, {7583.0, 7885.0},
    {15100.0, 17000.0}
};

static inline double gvh(int j, int i) {
    return (double)(float)(HSEGS[j][0] + (double)i * HSEGS[j][2]);
}

extern "C" void kernel_launch(void* const* d_in, const int* in_sizes, int n_in,
                              void* d_out, int out_size, void* d_ws, size_t ws_size,
                              hipStream_t stream) {
    if (n_in < 10) return;
    const float* z   = (const float*)d_in[0];
    const float* obs = (const float*)d_in[1];
    const float* W0  = (const float*)d_in[2];
    const float* W1  = (const float*)d_in[3];
    const float* W2  = (const float*)d_in[4];
    const float* W3  = (const float*)d_in[5];
    const float* b0  = (const float*)d_in[6];
    const float* b1  = (const float*)d_in[7];
    const float* b2  = (const float*)d_in[8];
    const float* b3  = (const float*)d_in[9];
    const int D = in_sizes[9];   // grid size (out_dim), authoritative

    // segment lengths exactly as np.arange: ceil((e+st - s)/st) in IEEE double
    int segN[7], segBase[7];
    int Dtot = 0;
    for (int j = 0; j < 7; ++j) {
        const double s = HSEGS[j][0], e = HSEGS[j][1], st = HSEGS[j][2];
        const int n = (int)ceil(((e + st) - s) / st);
        segBase[j] = Dtot; segN[j] = n; Dtot += n;
    }

    InterpParams P;
    for (int k = 0; k < 5; ++k) {
        IvInfo& iv = P.iv[k];
        const double s = HIV[k][0], e = HIV[k][1];
        iv.maskLo = (float)(s - 10.0);
        iv.maskHi = (float)(e + 10.0);
        iv.nSub = 0;
        for (int j = 0; j < 7 && iv.nSub < 3; ++j) {
            const double s0 = HSEGS[j][0], st = HSEGS[j][2];
            const int n = segN[j];
            // locA: first i with grid_f32 >= s ; locB: last i with grid_f32 <= e
            int a = (int)ceil((s - s0) / st);
            if (a < 0) a = 0; if (a > n - 1) a = n - 1;
            while (a > 0 && gvh(j, a - 1) >= s) --a;
            while (a < n && gvh(j, a) < s)      ++a;
            int bb = (int)floor((e - s0) / st);
            if (bb < 0) bb = 0; if (bb > n - 1) bb = n - 1;
            while (bb < n - 1 && gvh(j, bb + 1) <= e) ++bb;
            while (bb >= 0 && gvh(j, bb) > e)         --bb;
            if (a < n && bb >= 0 && a <= bb) {
                SubSeg& sg = iv.sub[iv.nSub++];
                sg.s0 = s0; sg.st = st; sg.base = segBase[j];
                sg.locA = a; sg.locB = bb;
            }
        }
        const SubSeg& f = iv.sub[0];
        const SubSeg& l = iv.sub[iv.nSub - 1];
        iv.iFirst = f.base + f.locA;
        iv.iLast  = l.base + l.locB;
        iv.gFirst = (float)(f.s0 + (double)f.locA * f.st);
        iv.gLast  = (float)(l.s0 + (double)l.locB * l.st);
    }

    // workspace layout (fp32): Y0 | Y1 | Y2 | spectra
    float* y0 = (float*)d_ws;
    float* y1 = y0 + (size_t)MBATCH * 512;
    float* y2 = y1 + (size_t)MBATCH * 2048;
    float* y3 = y2 + (size_t)MBATCH * 4096;

    const dim3 blk(256);
    gemm_wmma<true ><<<dim3(( 512 + BN - 1) / BN, MBATCH / BM), blk, 0, stream>>>(z,  W0, b0, y0, MBATCH,  512,  128);
    gemm_wmma<true ><<<dim3((2048 + BN - 1) / BN, MBATCH / BM), blk, 0, stream>>>(y0, W1, b1, y1, MBATCH, 2048,  512);
    gemm_wmma<true ><<<dim3((4096 + BN - 1) / BN, MBATCH / BM), blk, 0, stream>>>(y1, W2, b2, y2, MBATCH, 4096, 2048);
    gemm_wmma<false><<<dim3((D    + BN - 1) / BN, MBATCH / BM), blk, 0, stream>>>(y2, W3, b3, y3, MBATCH, D, 4096);

    const long total = (long)MBATCH * NOBS;
    interp_kernel<<<dim3((unsigned)((total + 255) / 256)), blk, 0, stream>>>(obs, y3, (float*)d_out, D, P);
}